// HyperNetwork_78159814852834
// MI455X (gfx1250) — compile-verified
//
#include <hip/hip_runtime.h>
#include <hip/hip_bf16.h>
#include <math.h>

#define DIM 256
#define HALF 128
#define TIME_DIM 1024
#define NEMB 2048
#define OUT_DIM (12 * NEMB * NEMB + 9 * NEMB)   // 50,350,080
#define NSTAGE 16                               // ceil(log2(OUT_DIM/TIME_DIM))
#define CHUNK 8192                              // final outputs per block
#define BW 4104                                 // >= CHUNK/2 + 8 (level-15 window)

typedef __attribute__((ext_vector_type(2))) float v2f;
typedef __attribute__((ext_vector_type(4))) float v4f;
typedef __attribute__((ext_vector_type(8))) float v8f;

// silu via hardware transcendentals: v_exp_f32 + v_rcp_f32 (~1 ulp), avoids
// the 11-op IEEE divide expansion that __fdividef produced.
__device__ __forceinline__ float silu_f(float x) {
    return x * __builtin_amdgcn_rcpf(1.0f + __expf(-x));
}

// ---------------------------------------------------------------------------
// Kernel 1: time-MLP. Single workgroup (32 waves).
//   e = sinusoidal(time)            (256)
//   h = silu(e @ w1^T + b1)         (1024)   -- WMMA f32 16x16x4
//   t = h @ w2^T + b2               (1024)   -- WMMA f32 16x16x4
// A fragment: row M=0 carries the activation vector (lanes l%16==0 hold K
// values, lane half selects K pair). B fragment: 4x16 tile of w^T, lane%16=N,
// (lane/16)*2+v = K. D row M=0 lives in accumulator VGPR0, lanes 0..15.
// Unroll capped at 2: A rebuilt from LDS per iteration (ds_load_2addr_b64 +
// cndmask) so nothing spills to scratch across the K loop.
// ---------------------------------------------------------------------------
__global__ void __launch_bounds__(1024)
time_mlp_kernel(const float* __restrict__ tim,
                const float* __restrict__ w1, const float* __restrict__ b1,
                const float* __restrict__ w2, const float* __restrict__ b2,
                float* __restrict__ t_out)
{
    __shared__ float e_lds[DIM];
    __shared__ float h_lds[TIME_DIM];
    const int tid = threadIdx.x;

    if (tid < DIM) {
        float tm   = tim[0];
        int   idx  = tid & (HALF - 1);
        float freq = expf((float)idx * (-logf(10000.0f) / (float)(HALF - 1)));
        float e    = tm * freq;
        e_lds[tid] = (tid < HALF) ? sinf(e) : cosf(e);
    }
    __syncthreads();

    const int wave = tid >> 5;
    const int lane = tid & 31;
    const int lrow = lane & 15;   // M for A-frag / N for B-frag
    const int lhi  = lane >> 4;   // K-half selector

    // layer 1: h = silu(e @ w1^T + b1), 64 tiles of 16 outputs
    for (int tile = wave; tile < TIME_DIM / 16; tile += 32) {
        const int n0 = tile * 16;
        v8f c = {};
        const float* wrow = w1 + (size_t)(n0 + lrow) * DIM;
#pragma unroll 2
        for (int kb = 0; kb < DIM; kb += 4) {
            int kk = kb + lhi * 2;
            v2f af, bf;
            float e0 = e_lds[kk], e1 = e_lds[kk + 1];
            af.x = (lrow == 0) ? e0 : 0.0f;
            af.y = (lrow == 0) ? e1 : 0.0f;
            bf.x = wrow[kk];
            bf.y = wrow[kk + 1];
            c = __builtin_amdgcn_wmma_f32_16x16x4_f32(false, af, false, bf,
                                                      (short)0, c, false, false);
        }
        if (lane < 16) {
            int n = n0 + lane;
            h_lds[n] = silu_f(c[0] + b1[n]);
        }
    }
    __syncthreads();

    // layer 2: t = h @ w2^T + b2
    for (int tile = wave; tile < TIME_DIM / 16; tile += 32) {
        const int n0 = tile * 16;
        v8f c = {};
        const float* wrow = w2 + (size_t)(n0 + lrow) * TIME_DIM;
#pragma unroll 2
        for (int kb = 0; kb < TIME_DIM; kb += 4) {
            int kk = kb + lhi * 2;
            v2f af, bf;
            float h0 = h_lds[kk], h1 = h_lds[kk + 1];
            af.x = (lrow == 0) ? h0 : 0.0f;
            af.y = (lrow == 0) ? h1 : 0.0f;
            bf.x = wrow[kk];
            bf.y = wrow[kk + 1];
            c = __builtin_amdgcn_wmma_f32_16x16x4_f32(false, af, false, bf,
                                                      (short)0, c, false, false);
        }
        if (lane < 16) {
            int n = n0 + lane;
            t_out[n] = c[0] + b2[n];
        }
    }
}

// ---------------------------------------------------------------------------
// Kernel 2: fused 16-stage transposed-conv expansion.
// Stage math (derived from jax conv_general_dilated with lhs_dilation=2,
// pad (2,2), kern=k[::-1]):  with s = silu(x), L' = 2L+1:
//   y[2m]   = k0*s[m] + k2*s[m-1] + b      (s out-of-range -> 0)
//   y[2m+1] = k1*s[m] + b
// Each block expands its CHUNK of final outputs level-by-level in LDS
// (ping-pong buffers); dependency window halves per level, so only ~4
// elements of t are read. Final level streamed to HBM via non-temporal
// float4 (b128) stores -- write-once data, keep it out of L2.
// ---------------------------------------------------------------------------
__global__ void __launch_bounds__(256)
expand_kernel(const float* __restrict__ t,
              const float* __restrict__ dk,
              const float* __restrict__ db,
              float* __restrict__ out)
{
    __shared__ float sA[BW];
    __shared__ float sB[BW];
    __shared__ float sdk[NSTAGE * 3];
    __shared__ float sdb[NSTAGE];

    const int tid = threadIdx.x;
    if (tid < NSTAGE * 3) sdk[tid] = dk[tid];
    if (tid >= 64 && tid < 64 + NSTAGE) sdb[tid - 64] = db[tid - 64];

    // level lengths: L[i] = 2*L[i-1] + 1
    int L[NSTAGE + 1];
    L[0] = TIME_DIM;
#pragma unroll
    for (int i = 1; i <= NSTAGE; ++i) L[i] = 2 * L[i - 1] + 1;

    // per-level dependency windows [a,b] (inclusive), fully unrolled -> SGPRs
    int a[NSTAGE + 1], b[NSTAGE + 1];
    a[NSTAGE] = blockIdx.x * CHUNK;
    {
        int e16 = a[NSTAGE] + CHUNK - 1;
        b[NSTAGE] = (e16 < OUT_DIM - 1) ? e16 : (OUT_DIM - 1);
    }
#pragma unroll
    for (int i = NSTAGE; i >= 1; --i) {
        int aa = (a[i] >> 1) - 1;
        a[i - 1] = (aa > 0) ? aa : 0;
        int bb  = b[i] >> 1;
        int lim = L[i - 1] - 1;
        b[i - 1] = (bb < lim) ? bb : lim;
    }

    // level 0: s0 = silu(t) into sA
    {
        int w0 = b[0] - a[0] + 1;
        for (int j = tid; j < w0; j += 256) sA[j] = silu_f(t[a[0] + j]);
    }
    __syncthreads();

    // levels 1..15: store silu(y) (consumed by next stage)
#pragma unroll
    for (int i = 1; i <= NSTAGE - 1; ++i) {
        const float* src = (i & 1) ? sA : sB;   // level i-1 lives in buf[(i-1)&1]
        float*       dst = (i & 1) ? sB : sA;
        const float k0 = sdk[(i - 1) * 3 + 0];
        const float k1 = sdk[(i - 1) * 3 + 1];
        const float k2 = sdk[(i - 1) * 3 + 2];
        const float bi = sdb[i - 1];
        const int lo = a[i - 1], hi = b[i - 1];
        const int wi = b[i] - a[i] + 1;
        for (int j = tid; j < wi; j += 256) {
            int n = a[i] + j;
            int m = n >> 1;
            float s0 = (m >= lo && m <= hi) ? src[m - lo] : 0.0f;
            float y;
            if (n & 1) {
                y = fmaf(k1, s0, bi);
            } else {
                float s1 = (m - 1 >= lo && m - 1 <= hi) ? src[m - 1 - lo] : 0.0f;
                y = fmaf(k0, s0, fmaf(k2, s1, bi));
            }
            dst[j] = silu_f(y);
        }
        __syncthreads();
    }

    // level 16: raw deconv output (no trailing silu), streamed with
    // non-temporal b128 stores
    {
        const float k0 = sdk[(NSTAGE - 1) * 3 + 0];
        const float k1 = sdk[(NSTAGE - 1) * 3 + 1];
        const float k2 = sdk[(NSTAGE - 1) * 3 + 2];
        const float bi = sdb[NSTAGE - 1];
        const float* src = sB;                  // level 15 in buf[15&1] = sB
        const int lo = a[NSTAGE - 1], hi = b[NSTAGE - 1];
        const int base = a[NSTAGE];
        const int cnt  = b[NSTAGE] - base + 1;
        const int cnt4 = cnt & ~3;
        for (int j = tid * 4; j < cnt4; j += 256 * 4) {
            int n = base + j;                   // base is even, so n is even
            int m = n >> 1;
            float sm1 = (m - 1 >= lo && m - 1 <= hi) ? src[m - 1 - lo] : 0.0f;
            float s0  = (m     >= lo && m     <= hi) ? src[m     - lo] : 0.0f;
            float s1  = (m + 1 >= lo && m + 1 <= hi) ? src[m + 1 - lo] : 0.0f;
            v4f y;
            y.x = fmaf(k0, s0, fmaf(k2, sm1, bi));
            y.y = fmaf(k1, s0, bi);
            y.z = fmaf(k0, s1, fmaf(k2, s0, bi));
            y.w = fmaf(k1, s1, bi);
            __builtin_nontemporal_store(y, reinterpret_cast<v4f*>(out + n));
        }
        for (int j = cnt4 + tid; j < cnt; j += 256) {
            int n = base + j;
            int m = n >> 1;
            float s0 = (m >= lo && m <= hi) ? src[m - lo] : 0.0f;
            float y;
            if (n & 1) {
                y = fmaf(k1, s0, bi);
            } else {
                float s1 = (m - 1 >= lo && m - 1 <= hi) ? src[m - 1 - lo] : 0.0f;
                y = fmaf(k0, s0, fmaf(k2, s1, bi));
            }
            __builtin_nontemporal_store(y, out + n);
        }
    }
}

extern "C" void kernel_launch(void* const* d_in, const int* in_sizes, int n_in,
                              void* d_out, int out_size, void* d_ws, size_t ws_size,
                              hipStream_t stream)
{
    const float* tim = (const float*)d_in[0];
    const float* w1  = (const float*)d_in[1];
    const float* b1  = (const float*)d_in[2];
    const float* w2  = (const float*)d_in[3];
    const float* b2  = (const float*)d_in[4];
    const float* dk  = (const float*)d_in[5];
    const float* db  = (const float*)d_in[6];

    float* t_ws = (float*)d_ws;   // 1024 f32 = 4 KB scratch for t

    time_mlp_kernel<<<1, 1024, 0, stream>>>(tim, w1, b1, w2, b2, t_ws);

    const int nblocks = (OUT_DIM + CHUNK - 1) / CHUNK;   // 6147
    expand_kernel<<<nblocks, 256, 0, stream>>>(t_ws, dk, db, (float*)d_out);
}